// SELF_supervised_47382079209801
// MI455X (gfx1250) — compile-verified
//
#include <hip/hip_runtime.h>

namespace {

constexpr int kN    = 50000;   // nodes
constexpr int kE    = 600000;  // edges
constexpr int kMask = 10000;   // masked nodes
constexpr int kNC   = 16;      // node-pred classes
constexpr float kEps = 1e-5f;

typedef __attribute__((ext_vector_type(2))) float v2f;
typedef __attribute__((ext_vector_type(8))) float v8f;

// ---------------------------------------------------------------------------
// out[M,Nc] = act( A[M,K] @ W1[Nc,K]^T  (+ A2[M,K] @ W2[Nc,K]^T) (+ bias[Nc]) )
// One wave computes one 16x16 output tile with V_WMMA_F32_16X16X4_F32.
// Requires M%16==0, Nc%16==0, K%4==0 (true for all call sites here).
// ---------------------------------------------------------------------------
__global__ void gemm_wmma_kernel(const float* __restrict__ A,
                                 const float* __restrict__ W1,
                                 const float* __restrict__ A2,
                                 const float* __restrict__ W2,
                                 const float* __restrict__ bias,
                                 float* __restrict__ out,
                                 int M, int K, int Nc, int do_relu)
{
    const int wavesPerBlock = blockDim.x >> 5;
    const int tile   = blockIdx.x * wavesPerBlock + (threadIdx.x >> 5);
    const int tilesN = Nc >> 4;
    const int tilesM = M  >> 4;
    if (tile >= tilesM * tilesN) return;           // wave-uniform

    const int tm = tile / tilesN;
    const int tn = tile - tm * tilesN;
    const int m0 = tm << 4;
    const int n0 = tn << 4;

    const int lane = threadIdx.x & 31;
    const int hi   = lane >> 4;                    // wave half: 0 or 1
    const int l    = lane & 15;

    // C/D layout: VGPR r, lanes<16 -> (M=r, N=lane); lanes>=16 -> (M=8+r, N=lane-16)
    v8f acc;
    const float bcol = bias ? bias[n0 + l] : 0.0f;
#pragma unroll
    for (int r = 0; r < 8; ++r) acc[r] = bcol;

    const float* arow = A  + (size_t)(m0 + l) * K;   // A row M=l (both halves)
    const float* wrow = W1 + (size_t)(n0 + l) * K;   // B[k][n] = W1[n][k]
    for (int kk = 0; kk < K; kk += 4) {              // uniform trip count
        const int k = kk + 2 * hi;                   // half0: K=kk,kk+1; half1: K=kk+2,kk+3
        v2f av, bv;
        av.x = arow[k]; av.y = arow[k + 1];
        bv.x = wrow[k]; bv.y = wrow[k + 1];
        acc = __builtin_amdgcn_wmma_f32_16x16x4_f32(
            false, av, false, bv, (short)0, acc, false, false);
    }
    if (A2) {
        const float* arow2 = A2 + (size_t)(m0 + l) * K;
        const float* wrow2 = W2 + (size_t)(n0 + l) * K;
        for (int kk = 0; kk < K; kk += 4) {
            const int k = kk + 2 * hi;
            v2f av, bv;
            av.x = arow2[k]; av.y = arow2[k + 1];
            bv.x = wrow2[k]; bv.y = wrow2[k + 1];
            acc = __builtin_amdgcn_wmma_f32_16x16x4_f32(
                false, av, false, bv, (short)0, acc, false, false);
        }
    }
#pragma unroll
    for (int r = 0; r < 8; ++r) {
        float v = acc[r];
        if (do_relu) v = fmaxf(v, 0.0f);
        out[(size_t)(m0 + hi * 8 + r) * Nc + n0 + l] = v;
    }
}

// ---------------------------------------------------------------------------
__global__ void fill_kernel(float* __restrict__ p, int n, float v)
{
    int i = blockIdx.x * blockDim.x + threadIdx.x;
    if (i < n) p[i] = v;
}

__global__ void copy_kernel(const float* __restrict__ src, float* __restrict__ dst, int n)
{
    int i = blockIdx.x * blockDim.x + threadIdx.x;
    if (i < n) dst[i] = src[i];
}

// overlay mask token rows: h[mask_nodes] = token
__global__ void mask_set_kernel(const int* __restrict__ mask_nodes,
                                const float* __restrict__ token,
                                float* __restrict__ h)
{
    int i = blockIdx.x * blockDim.x + threadIdx.x;
    if (i < kMask * 128) {
        int node = mask_nodes[i >> 7];
        int f    = i & 127;
        h[(size_t)node * 128 + f] = token[f];
    }
}

// neigh[dst[e],f] = max(neigh[dst[e],f], p[src[e],f]*ew[e]); values >= 0 so
// signed-int atomicMax on the float bit pattern is exact.
__global__ void edge_max_kernel(const float* __restrict__ p,
                                const int* __restrict__ src,
                                const int* __restrict__ dst,
                                const float* __restrict__ ew,
                                float* __restrict__ neigh,
                                int din)
{
    int idx = blockIdx.x * blockDim.x + threadIdx.x;
    int total = kE * din;                 // <= 76.8M, fits int
    if (idx >= total) return;
    int e = idx / din;
    int f = idx - e * din;
    float val = p[(size_t)src[e] * din + f] * ew[e];
    atomicMax((int*)&neigh[(size_t)dst[e] * din + f], __float_as_int(val));
}

// LayerNorm over 16 features per node
__global__ void nodepred_ln_kernel(const float* __restrict__ z,
                                   const float* __restrict__ g,
                                   const float* __restrict__ b,
                                   float* __restrict__ out)
{
    int i = blockIdx.x * blockDim.x + threadIdx.x;
    if (i >= kN) return;
    const float* zr = z + (size_t)i * kNC;
    float v[kNC];
    float mu = 0.0f;
#pragma unroll
    for (int c = 0; c < kNC; ++c) { v[c] = zr[c]; mu += v[c]; }
    mu *= (1.0f / kNC);
    float var = 0.0f;
#pragma unroll
    for (int c = 0; c < kNC; ++c) { float d = v[c] - mu; var += d * d; }
    var *= (1.0f / kNC);
    float inv = rsqrtf(var + kEps);
#pragma unroll
    for (int c = 0; c < kNC; ++c)
        out[(size_t)i * kNC + c] = (v[c] - mu) * inv * g[c] + b[c];
}

__global__ void gather_rows_kernel(const float* __restrict__ srcbuf,
                                   const int* __restrict__ idx,
                                   float* __restrict__ dstbuf,
                                   int nrows, int width)
{
    int i = blockIdx.x * blockDim.x + threadIdx.x;
    if (i < nrows * width) {
        int r = i / width;
        int f = i - r * width;
        dstbuf[i] = srcbuf[(size_t)idx[r] * width + f];
    }
}

} // namespace

// ---------------------------------------------------------------------------
extern "C" void kernel_launch(void* const* d_in, const int* in_sizes, int n_in,
                              void* d_out, int out_size, void* d_ws, size_t ws_size,
                              hipStream_t stream)
{
    const float* x     = (const float*)d_in[0];
    const int*   src   = (const int*)  d_in[1];
    const int*   dst   = (const int*)  d_in[2];
    const float* ew    = (const float*)d_in[3];
    const int*   mask  = (const int*)  d_in[4];
    const float* token = (const float*)d_in[5];

    // per-layer params: Wp, bp, Ws, Wn, b at d_in[6 + 5*layer + j]
    const float* P[4][5];
    for (int L = 0; L < 4; ++L)
        for (int j = 0; j < 5; ++j)
            P[L][j] = (const float*)d_in[6 + 5 * L + j];
    const float* W_e2d = (const float*)d_in[26];
    const float* Wnp   = (const float*)d_in[27];
    const float* bnp   = (const float*)d_in[28];
    const float* ln_g  = (const float*)d_in[29];
    const float* ln_b  = (const float*)d_in[30];

    float* A = (float*)d_ws;                       // 50000*128 f32 = 25.6 MB
    float* B = A + (size_t)kN * 128;
    float* C = B + (size_t)kN * 128;               // neigh scratch
    float* Z = C + (size_t)kN * 128;               // node-pred logits (50000*16)

    float* out_r  = (float*)d_out;                       // r[mask]  : 10000*128
    float* out_x  = out_r + (size_t)kMask * 128;         // x[mask]  : 10000*128
    float* out_ns = out_x + (size_t)kMask * 128;         // n_scores : 50000*16

    auto gemm = [&](const float* Ain, const float* W1, const float* A2,
                    const float* W2, const float* bias, float* outp,
                    int M, int K, int Nc, int relu) {
        int tiles  = (M / 16) * (Nc / 16);
        int blocks = (tiles + 7) / 8;                    // 8 waves / 256-thread block
        gemm_wmma_kernel<<<blocks, 256, 0, stream>>>(Ain, W1, A2, W2, bias, outp,
                                                     M, K, Nc, relu);
    };

    // One SAGE 'pool' layer: hin -> Y (Y also used as scratch for p)
    auto sage = [&](const float* hin, float* Y, int din, int dout, const float* const* prm) {
        gemm(hin, prm[0], nullptr, nullptr, prm[1], Y, kN, din, din, 1);  // p = relu(h@Wp^T+bp)
        int nd = kN * din;
        fill_kernel<<<(nd + 255) / 256, 256, 0, stream>>>(C, nd, 0.0f);
        int et = kE * din;
        edge_max_kernel<<<(et + 255) / 256, 256, 0, stream>>>(Y, src, dst, ew, C, din);
        gemm(hin, prm[2], C, prm[3], prm[4], Y, kN, din, dout, 1);        // relu(h@Ws^T+neigh@Wn^T+b)
    };

    // h0 = x with masked rows replaced by token
    copy_kernel<<<(kN * 128 + 255) / 256, 256, 0, stream>>>(x, A, kN * 128);
    mask_set_kernel<<<(kMask * 128 + 255) / 256, 256, 0, stream>>>(mask, token, A);

    sage(A, B, 128, 96, P[0]);     // enc0: h1 in B
    sage(B, A,  96, 64, P[1]);     // enc1: h2 in A

    // node-pred head + LayerNorm
    gemm(A, Wnp, nullptr, nullptr, bnp, Z, kN, 64, kNC, 0);
    nodepred_ln_kernel<<<(kN + 255) / 256, 256, 0, stream>>>(Z, ln_g, ln_b, out_ns);

    // encoder_to_decoder (no bias, no relu): r0 in B
    gemm(A, W_e2d, nullptr, nullptr, nullptr, B, kN, 64, 64, 0);

    sage(B, A,  64,  96, P[2]);    // dec0: r1 in A
    sage(A, B,  96, 128, P[3]);    // dec1: r  in B

    gather_rows_kernel<<<(kMask * 128 + 255) / 256, 256, 0, stream>>>(B, mask, out_r, kMask, 128);
    gather_rows_kernel<<<(kMask * 128 + 255) / 256, 256, 0, stream>>>(x, mask, out_x, kMask, 128);
}